// GraphBlock_9629316678019
// MI455X (gfx1250) — compile-verified
//
#include <hip/hip_runtime.h>
#include <hip/hip_bf16.h>
#include <math.h>

typedef __attribute__((ext_vector_type(2))) float v2f;
typedef __attribute__((ext_vector_type(8))) float v8f;

__device__ __forceinline__ v8f wmma4(v2f a, v2f b, v8f c) {
  // D = A(16x4) * B(4x16) + C, f32. (neg_a, A, neg_b, B, c_mod, C, reuse_a, reuse_b)
  return __builtin_amdgcn_wmma_f32_16x16x4_f32(false, a, false, b, (short)0, c, false, false);
}

// ---------------- problem constants -----------------
#define NB 32
#define LL 1024
#define HH 512
#define DD 1024
#define PP 10
#define SS 60
#define EN 80
#define EE 151
#define EP 160          // padded node count (multiple of 32, mult of 4 for K)
#define TD 2048

// output offsets (floats, concatenated in reference return order)
#define O_COPY  0
#define O_GAT   33554432
#define O_NM    38502400
#define O_SENT  38507232
#define O_QOUT  40473312
#define O_PLOG  40506080
#define O_PPRED 40506400
#define O_SLOG  40507040
#define O_SPRED 40508960
#define O_ELOG  40512800

// workspace offsets (floats)
#define W_X   0                 // graph_state padded (NB,EP,DD)
#define W_H   5242880           // h matrix padded   (NB,EP,DD)
#define W_CT  10485760          // coefs transposed  (NB,EP,EP)
#define W_HID 11304960          // hidden buffer     (2560,TD) reused

// ---------------- helpers -----------------
__device__ __forceinline__ float nmask(int n, int e,
    const float* pm, const float* sm, const float* em) {
  if (e == 0)  return 1.f;
  if (e < 11)  return pm[n*PP + e - 1];
  if (e < 71)  return sm[n*SS + e - 11];
  if (e < EE)  return em[n*EN + e - 71];
  return 0.f;
}

// ---------------- simple kernels -----------------
__global__ void k_zero(float* p, int n) {
  int i = blockIdx.x*blockDim.x + threadIdx.x;
  if (i < n) p[i] = 0.f;
}

__global__ void k_zero_xpad(float* X) {
  int i = blockIdx.x*blockDim.x + threadIdx.x;   // [0, 32*9*1024)
  int n = i / (9*1024); int rem = i % (9*1024);
  int e = 151 + rem/1024; int c = rem % 1024;
  X[(n*EP + e)*DD + c] = 0.f;
}

__global__ void k_copy4(const float4* __restrict__ in, float4* __restrict__ out) {
  int i = blockIdx.x*blockDim.x + threadIdx.x;
  out[i] = in[i];
}

// gs0 = query_vec @ gat_linear_w + b  -> graph_state node 0
__global__ void k_gs0(const float* __restrict__ q, const float* __restrict__ w,
                      const float* __restrict__ b, float* __restrict__ X) {
  __shared__ float qs[DD];
  int n = blockIdx.y;
  int col = blockIdx.x*256 + threadIdx.x;
  for (int i = threadIdx.x; i < DD; i += 256) qs[i] = q[n*DD + i];
  __syncthreads();
  float acc = b[col];
  for (int d = 0; d < DD; ++d) acc += qs[d] * w[d*DD + col];
  X[(n*EP + 0)*DD + col] = acc;
}

// para/sent/ent state: weighted gather over L (skips zero map entries; exact for any map)
__global__ void k_gather(const float* __restrict__ inp,
    const float* __restrict__ psm, const float* __restrict__ pem,
    const float* __restrict__ ssm, const float* __restrict__ sem,
    const float* __restrict__ esm, const float* __restrict__ eem,
    float* __restrict__ X, float* __restrict__ sentOut) {
  __shared__ float wrow[LL];
  int b = blockIdx.x;
  int hf = b & 1; int tmp = b >> 1;
  int node = tmp % 150; int n = tmp / 150;
  int e = node + 1;
  const float* map;
  if (e < 11)      map = (hf ? pem : psm) + (n*PP + (e-1))*LL;
  else if (e < 71) map = (hf ? sem : ssm) + (n*SS + (e-11))*LL;
  else             map = (hf ? eem : esm) + (n*EN + (e-71))*LL;
  int t = threadIdx.x;                  // 0..511
  wrow[t] = map[t]; wrow[t+512] = map[t+512];
  __syncthreads();
  // start-map (hf==0) reads input[:, :, H:]; end-map (hf==1) reads input[:, :, :H]
  const float* src = inp + n*LL*DD + (hf ? 0 : HH) + t;
  float acc = 0.f;
  for (int l = 0; l < LL; ++l) {
    float wv = wrow[l];
    if (wv != 0.f) acc += wv * src[l*DD];   // uniform branch per block
  }
  X[(n*EP + e)*DD + hf*HH + t] = acc;
  if (e >= 11 && e < 71) sentOut[(n*SS + (e-11))*DD + hf*HH + t] = acc;
}

// coefs = softmax(where(g>0,0,-1e30), axis=j) == (g>0)/cnt  (uniform 1/E if cnt==0)
// stored transposed & zero-padded: CT[n, j, i]
__global__ void k_coef(const float* __restrict__ graphs, float* __restrict__ CT) {
  __shared__ float red[256];
  int b = blockIdx.x; int n = b / EE; int i = b % EE;
  int j = threadIdx.x;
  float g = (j < EE) ? graphs[(n*EE + i)*EE + j] : 0.f;
  float pos = (g > 0.f) ? 1.f : 0.f;
  red[j] = (j < EE) ? pos : 0.f;
  __syncthreads();
  for (int s = 128; s > 0; s >>= 1) { if (j < s) red[j] += red[j+s]; __syncthreads(); }
  float cnt = red[0];
  __syncthreads();
  if (j < EE) {
    float val = (cnt > 0.f) ? (pos / cnt) : (1.f / (float)EE);
    CT[n*EP*EP + j*EP + i] = val;
  }
}

__global__ void k_nodemask(const float* pm, const float* sm, const float* em, float* out) {
  int i = blockIdx.x*blockDim.x + threadIdx.x;
  if (i >= NB*EE) return;
  int n = i / EE, e = i % EE;
  out[i] = nmask(n, e, pm, sm, em);
}

// ---------------- WMMA GEMM kernels (2x2 register-blocked: 32x32 per wave) -----------------
// h = X(5120x1024) @ W' (1024x1024), W'[d, hd*128+k] = gat_W[hd,d,k]; row-scaled by node mask
__global__ void k_gemm_h(const float* __restrict__ X, const float* __restrict__ gatW,
                         const float* pm, const float* sm, const float* em,
                         float* __restrict__ Hd) {
  int lane = threadIdx.x & 31, wv = threadIdx.x >> 5;
  int ntp = blockIdx.x*4 + wv;          // 0..31 (pair of n-tiles)
  int mtp = blockIdx.y;                 // 0..159 (pair of m-tiles)
  int kh = lane >> 4, mr = lane & 15;
  int m0 = mtp*32;
  int col0 = ntp*32 + mr, col1 = col0 + 16;
  const float* aRow0 = X + (m0 + mr)*DD;
  const float* aRow1 = aRow0 + 16*DD;
  const float* bCol0 = gatW + ((col0 >> 7) << 17) + (col0 & 127);  // + d*128
  const float* bCol1 = gatW + ((col1 >> 7) << 17) + (col1 & 127);
  v8f a00 = {}, a01 = {}, a10 = {}, a11 = {};
  #pragma unroll 2
  for (int k0 = 0; k0 < DD; k0 += 4) {
    int kk = k0 + 2*kh;
    v2f a0, a1, b0, b1;
    a0[0] = aRow0[kk];      a0[1] = aRow0[kk+1];
    a1[0] = aRow1[kk];      a1[1] = aRow1[kk+1];
    b0[0] = bCol0[kk << 7]; b0[1] = bCol0[(kk+1) << 7];
    b1[0] = bCol1[kk << 7]; b1[1] = bCol1[(kk+1) << 7];
    a00 = wmma4(a0, b0, a00);
    a01 = wmma4(a0, b1, a01);
    a10 = wmma4(a1, b0, a10);
    a11 = wmma4(a1, b1, a11);
  }
  int n = m0 / EP, e0 = m0 % EP;
  int rbase = kh ? 8 : 0;
  #pragma unroll
  for (int g = 0; g < 8; ++g) {
    float mv0 = nmask(n, e0 + rbase + g,      pm, sm, em);
    float mv1 = nmask(n, e0 + 16 + rbase + g, pm, sm, em);
    Hd[(m0 + rbase + g)*DD + col0]      = a00[g] * mv0;
    Hd[(m0 + rbase + g)*DD + col1]      = a01[g] * mv0;
    Hd[(m0 + 16 + rbase + g)*DD + col0] = a10[g] * mv1;
    Hd[(m0 + 16 + rbase + g)*DD + col1] = a11[g] * mv1;
  }
}

// agg = coefs^T(160x160) @ h(160x1024) per batch; relu -> gat_out; row0 -> query_out
__global__ void k_gemm_agg(const float* __restrict__ CT, const float* __restrict__ Hd,
                           float* __restrict__ gat, float* __restrict__ qout) {
  int lane = threadIdx.x & 31, wv = threadIdx.x >> 5;
  int ntp = blockIdx.x*4 + wv;          // 0..31
  int mtp = blockIdx.y;                 // 0..4
  int n = blockIdx.z;                   // 0..31
  int kh = lane >> 4, mr = lane & 15;
  int m0 = mtp*32;
  int col0 = ntp*32 + mr, col1 = col0 + 16;
  const float* aRow0 = CT + n*EP*EP + (m0 + mr)*EP;
  const float* aRow1 = aRow0 + 16*EP;
  const float* bBase = Hd + n*EP*DD;
  v8f a00 = {}, a01 = {}, a10 = {}, a11 = {};
  #pragma unroll 2
  for (int k0 = 0; k0 < EP; k0 += 4) {
    int kk = k0 + 2*kh;
    v2f a0, a1, b0, b1;
    a0[0] = aRow0[kk];            a0[1] = aRow0[kk+1];
    a1[0] = aRow1[kk];            a1[1] = aRow1[kk+1];
    b0[0] = bBase[kk*DD + col0];  b0[1] = bBase[(kk+1)*DD + col0];
    b1[0] = bBase[kk*DD + col1];  b1[1] = bBase[(kk+1)*DD + col1];
    a00 = wmma4(a0, b0, a00);
    a01 = wmma4(a0, b1, a01);
    a10 = wmma4(a1, b0, a10);
    a11 = wmma4(a1, b1, a11);
  }
  int rbase = kh ? 8 : 0;
  #pragma unroll
  for (int g = 0; g < 8; ++g) {
    int j0 = m0 + rbase + g, j1 = j0 + 16;
    if (j0 < EE) {
      float v0 = fmaxf(a00[g], 0.f), v1 = fmaxf(a01[g], 0.f);
      gat[(n*EE + j0)*DD + col0] = v0;
      gat[(n*EE + j0)*DD + col1] = v1;
      if (j0 == 0) { qout[n*DD + col0] = v0; qout[n*DD + col1] = v1; }
    }
    if (j1 < EE) {
      gat[(n*EE + j1)*DD + col0] = fmaxf(a10[g], 0.f);
      gat[(n*EE + j1)*DD + col1] = fmaxf(a11[g], 0.f);
    }
  }
}

// hidden = relu(x @ w1 + b1); x rows gathered from gat_out node slices
__global__ void k_gemm_w1(const float* __restrict__ gat, const float* __restrict__ w1,
                          const float* __restrict__ b1, float* __restrict__ hid,
                          int rowsPer, int eBase) {
  int lane = threadIdx.x & 31, wv = threadIdx.x >> 5;
  int ntp = blockIdx.x*4 + wv;          // 0..63 (pair of n-tiles over 2048)
  int mtp = blockIdx.y;
  int kh = lane >> 4, mr = lane & 15;
  int m0 = mtp*32;
  int col0 = ntp*32 + mr, col1 = col0 + 16;
  int r0 = m0 + mr,      r1 = r0 + 16;
  int nb0 = r0 / rowsPer, e0 = eBase + r0 % rowsPer;
  int nb1 = r1 / rowsPer, e1 = eBase + r1 % rowsPer;
  const float* aRow0 = gat + (nb0*EE + e0)*DD;
  const float* aRow1 = gat + (nb1*EE + e1)*DD;
  const float* bCol0 = w1 + col0;
  const float* bCol1 = w1 + col1;
  v8f a00 = {}, a01 = {}, a10 = {}, a11 = {};
  #pragma unroll 2
  for (int k0 = 0; k0 < DD; k0 += 4) {
    int kk = k0 + 2*kh;
    v2f a0, a1, b0, b1;
    a0[0] = aRow0[kk];       a0[1] = aRow0[kk+1];
    a1[0] = aRow1[kk];       a1[1] = aRow1[kk+1];
    b0[0] = bCol0[kk*TD];    b0[1] = bCol0[(kk+1)*TD];
    b1[0] = bCol1[kk*TD];    b1[1] = bCol1[(kk+1)*TD];
    a00 = wmma4(a0, b0, a00);
    a01 = wmma4(a0, b1, a01);
    a10 = wmma4(a1, b0, a10);
    a11 = wmma4(a1, b1, a11);
  }
  float bias0 = b1[col0], bias1 = b1[col1];
  int rbase = kh ? 8 : 0;
  #pragma unroll
  for (int g = 0; g < 8; ++g) {
    hid[(m0 + rbase + g)*TD + col0]      = fmaxf(a00[g] + bias0, 0.f);
    hid[(m0 + rbase + g)*TD + col1]      = fmaxf(a01[g] + bias1, 0.f);
    hid[(m0 + 16 + rbase + g)*TD + col0] = fmaxf(a10[g] + bias0, 0.f);
    hid[(m0 + 16 + rbase + g)*TD + col1] = fmaxf(a11[g] + bias1, 0.f);
  }
}

// layernorm over 2048 + dot(w2) + b2, then scatter logits per mode
__global__ void k_ln_logit(const float* __restrict__ hid,
    const float* __restrict__ lnw, const float* __restrict__ lnb,
    const float* __restrict__ w2, const float* __restrict__ b2,
    float* plog, float* ppred, float* slog, float* spred,
    float* elog, const float* acm, int mode) {
  __shared__ float red[256];
  int r = blockIdx.x, t = threadIdx.x;
  const float* row = hid + r*TD;
  float v[8];
  float s0 = 0.f;
  #pragma unroll
  for (int i = 0; i < 8; ++i) { v[i] = row[t + 256*i]; s0 += v[i]; }
  red[t] = s0; __syncthreads();
  for (int s = 128; s > 0; s >>= 1) { if (t < s) red[t] += red[t+s]; __syncthreads(); }
  float u = red[0] * (1.f/TD); __syncthreads();
  float s1 = 0.f;
  #pragma unroll
  for (int i = 0; i < 8; ++i) { float x = v[i]-u; s1 += x*x; }
  red[t] = s1; __syncthreads();
  for (int s = 128; s > 0; s >>= 1) { if (t < s) red[t] += red[t+s]; __syncthreads(); }
  float inv = 1.f / sqrtf(red[0]*(1.f/TD) + 1e-12f); __syncthreads();
  float s2 = 0.f;
  #pragma unroll
  for (int i = 0; i < 8; ++i) {
    int c = t + 256*i;
    s2 += (lnw[c]*(v[i]-u)*inv + lnb[c]) * w2[c];
  }
  red[t] = s2; __syncthreads();
  for (int s = 128; s > 0; s >>= 1) { if (t < s) red[t] += red[t+s]; __syncthreads(); }
  if (t == 0) {
    float logit = red[0] + b2[0];
    if (mode == 0) {
      int n = r / 71, e = r % 71;
      if (e >= 1 && e < 11) {
        int p = n*PP + (e-1);
        plog[p] = logit; ppred[2*p] = 0.f; ppred[2*p+1] = logit;
      } else if (e >= 11) {
        int si = n*SS + (e-11);
        slog[si] = logit; spred[2*si] = 0.f; spred[2*si+1] = logit;
      }
    } else {
      elog[r] = logit - 1e30f*(1.f - acm[r]);
    }
  }
}

extern "C" void kernel_launch(void* const* d_in, const int* in_sizes, int n_in,
                              void* d_out, int out_size, void* d_ws, size_t ws_size,
                              hipStream_t stream) {
  (void)in_sizes; (void)n_in; (void)out_size; (void)ws_size;
  const float* inp    = (const float*)d_in[0];
  const float* q      = (const float*)d_in[1];
  const float* psm    = (const float*)d_in[2];
  const float* pem    = (const float*)d_in[3];
  const float* ssm    = (const float*)d_in[4];
  const float* sem    = (const float*)d_in[5];
  const float* esm    = (const float*)d_in[6];
  const float* eem    = (const float*)d_in[7];
  const float* pmask  = (const float*)d_in[8];
  const float* smask  = (const float*)d_in[9];
  const float* emask  = (const float*)d_in[10];
  const float* graphs = (const float*)d_in[11];
  const float* acm    = (const float*)d_in[12];
  const float* glw    = (const float*)d_in[13];
  const float* glb    = (const float*)d_in[14];
  const float* gatW   = (const float*)d_in[15];
  const float* sw1    = (const float*)d_in[16];
  const float* sb1    = (const float*)d_in[17];
  const float* slnw   = (const float*)d_in[18];
  const float* slnb   = (const float*)d_in[19];
  const float* sw2    = (const float*)d_in[20];
  const float* sb2    = (const float*)d_in[21];
  const float* ew1    = (const float*)d_in[22];
  const float* eb1    = (const float*)d_in[23];
  const float* elnw   = (const float*)d_in[24];
  const float* elnb   = (const float*)d_in[25];
  const float* ew2    = (const float*)d_in[26];
  const float* eb2    = (const float*)d_in[27];

  float* out = (float*)d_out;
  float* ws  = (float*)d_ws;
  float* X   = ws + W_X;
  float* Hd  = ws + W_H;
  float* CT  = ws + W_CT;
  float* HID = ws + W_HID;

  k_zero_xpad<<<(NB*9*1024)/256, 256, 0, stream>>>(X);
  k_zero<<<(NB*EP*EP + 255)/256, 256, 0, stream>>>(CT, NB*EP*EP);
  k_copy4<<<(NB*LL*DD/4)/256, 256, 0, stream>>>((const float4*)inp, (float4*)(out + O_COPY));
  k_gs0<<<dim3(4, NB), 256, 0, stream>>>(q, glw, glb, X);
  k_gather<<<NB*150*2, 512, 0, stream>>>(inp, psm, pem, ssm, sem, esm, eem, X, out + O_SENT);
  k_coef<<<NB*EE, 256, 0, stream>>>(graphs, CT);
  k_nodemask<<<(NB*EE + 255)/256, 256, 0, stream>>>(pmask, smask, emask, out + O_NM);
  k_gemm_h<<<dim3(8, (NB*EP)/32), 128, 0, stream>>>(X, gatW, pmask, smask, emask, Hd);
  k_gemm_agg<<<dim3(8, EP/32, NB), 128, 0, stream>>>(CT, Hd, out + O_GAT, out + O_QOUT);
  k_gemm_w1<<<dim3(16, (NB*71)/32), 128, 0, stream>>>(out + O_GAT, sw1, sb1, HID, 71, 0);
  k_ln_logit<<<NB*71, 256, 0, stream>>>(HID, slnw, slnb, sw2, sb2,
      out + O_PLOG, out + O_PPRED, out + O_SLOG, out + O_SPRED, nullptr, nullptr, 0);
  k_gemm_w1<<<dim3(16, (NB*EN)/32), 128, 0, stream>>>(out + O_GAT, ew1, eb1, HID, EN, 71);
  k_ln_logit<<<NB*EN, 256, 0, stream>>>(HID, elnw, elnb, ew2, eb2,
      nullptr, nullptr, nullptr, nullptr, out + O_ELOG, acm, 1);
}